// SelfAttention_38895223832932
// MI455X (gfx1250) — compile-verified
//
#include <hip/hip_runtime.h>
#include <cstdint>

// CDNA5 fp32 WMMA fragments: A/B = 2 VGPRs, C/D = 8 VGPRs (wave32)
typedef float v2f __attribute__((ext_vector_type(2)));
typedef float v8f __attribute__((ext_vector_type(8)));

#define N_BATCH 32
#define SEQ     1024
#define EMBED   128
#define HEADS   4
#define HD      32
#define SCALE   0.08838834764831843f   // 1/sqrt(EMBED) -- faithful to reference

// ---------------------------------------------------------------------------
// Kernel A: fused per-head q/k/v projections.  q[n,h,s,:] = Wq @ x[n,s,h*32:...]
// Output layout [N,H,S,D] so each (n,h) slice is a contiguous S x 32 matrix.
// ---------------------------------------------------------------------------
__global__ void qkv_proj_kernel(const float* __restrict__ values,
                                const float* __restrict__ keys,
                                const float* __restrict__ query,
                                const float* __restrict__ Wv,
                                const float* __restrict__ Wk,
                                const float* __restrict__ Wq,
                                float* __restrict__ qw,
                                float* __restrict__ kw,
                                float* __restrict__ vw)
{
    __shared__ float sW[3 * 1024];   // Wq | Wk | Wv, 32x32 row-major each
    __shared__ float sx[3 * 128];    // this token's query | keys | values rows
    const int tid   = threadIdx.x;   // 128 threads
    const int token = blockIdx.x;    // n*SEQ + s
    const int n     = token >> 10;
    const int s     = token & (SEQ - 1);

    for (int i = tid; i < 1024; i += 128) {
        sW[i]        = Wq[i];
        sW[1024 + i] = Wk[i];
        sW[2048 + i] = Wv[i];
    }
    const size_t tb = (size_t)token * EMBED;
    sx[tid]       = query[tb + tid];
    sx[128 + tid] = keys[tb + tid];
    sx[256 + tid] = values[tb + tid];
    __syncthreads();

    const int h = tid >> 5;
    const int i = tid & 31;
    float aq = 0.f, ak = 0.f, av = 0.f;
#pragma unroll
    for (int j = 0; j < 32; ++j) {
        aq += sW[i * 32 + j]        * sx[h * 32 + j];
        ak += sW[1024 + i * 32 + j] * sx[128 + h * 32 + j];
        av += sW[2048 + i * 32 + j] * sx[256 + h * 32 + j];
    }
    const size_t o = ((size_t)(n * HEADS + h) * SEQ + s) * HD + i;
    qw[o] = aq; kw[o] = ak; vw[o] = av;
}

// ---------------------------------------------------------------------------
// Kernel B: per (n, h, 16-row tile):
//   phase1: scores(16x1024) = Q(16x32) @ K^T   via v_wmma_f32_16x16x4_f32
//   phase2: mask + scale + softmax (shfl reductions), stream weights to HBM
//           (mask loads + attw stores use NT temporal hints so the 64x-reused
//            K/V/Q slices stay resident in the 192 MB L2)
//   phase3: head_out(16x32) = attw(16x1024) @ V via WMMA, ds_add_f32 merge
// Fragment layout assumptions (ISA 7.12.2):
//   A 16x4 : lane l holds row M=l%16; VGPR v holds K = 2*(l>=16) + v
//   B 4x16 : lane l holds col N=l%16; VGPR v holds K = 2*(l>=16) + v (mirror of A)
//   C 16x16: VGPR r holds rows M=r (lanes 0-15) and M=r+8 (lanes 16-31), N=l%16
// ---------------------------------------------------------------------------
__global__ void attention_kernel(const float* __restrict__ qw,
                                 const float* __restrict__ kw,
                                 const float* __restrict__ vw,
                                 const int*   __restrict__ mask,
                                 float* __restrict__ attw,
                                 float* __restrict__ attout)
{
    extern __shared__ float smem[];
    float* scores = smem;             // [16][SEQ]   64 KB
    float* accum  = smem + 16 * SEQ;  // [16][HD]    2 KB

    const int tid   = threadIdx.x;    // 256 threads = 8 waves
    const int lane  = tid & 31;
    const int wv    = tid >> 5;
    const int row0  = blockIdx.x * 16;
    const int h     = blockIdx.y;
    const int n     = blockIdx.z;
    const int lm    = lane & 15;
    const int khalf = (lane >> 4) << 1;   // 0 (lanes 0-15) or 2 (lanes 16-31)

    accum[tid] = 0.f; accum[tid + 256] = 0.f;

    const size_t headBase = (size_t)(n * HEADS + h) * SEQ * HD;
    const float* qb = qw + headBase + (size_t)row0 * HD;
    const float* kb = kw + headBase;
    const float* vb = vw + headBase;

    // ---- phase 1: score tiles ----
    v2f a[8];
#pragma unroll
    for (int kc = 0; kc < 8; ++kc) {
        const int kk = kc * 4 + khalf;
        a[kc].x = qb[lm * HD + kk];
        a[kc].y = qb[lm * HD + kk + 1];
    }
    for (int jt = wv; jt < SEQ / 16; jt += 8) {
        const int col0 = jt * 16;
        v8f c = {};
#pragma unroll
        for (int kc = 0; kc < 8; ++kc) {
            const int kk = kc * 4 + khalf;
            v2f b;
            b.x = kb[(size_t)(col0 + lm) * HD + kk];
            b.y = kb[(size_t)(col0 + lm) * HD + kk + 1];
            c = __builtin_amdgcn_wmma_f32_16x16x4_f32(false, a[kc], false, b,
                                                      (short)0, c, false, false);
        }
#pragma unroll
        for (int r = 0; r < 8; ++r)
            scores[(r + (khalf << 2)) * SEQ + col0 + lm] = c[r];
    }
    __syncthreads();

    // ---- phase 2: mask + scale + softmax; 2 rows per wave ----
#pragma unroll
    for (int rr = 0; rr < 2; ++rr) {
        const int r    = wv * 2 + rr;
        const int grow = row0 + r;
        const int* mrow = mask + ((size_t)n * SEQ + grow) * SEQ;   // mask[n,0,row,:]
        float* srow = scores + r * SEQ;

        float mx = -3.4e38f;
        for (int cI = lane; cI < SEQ; cI += 32) {
            float t = srow[cI];
            const int m = __builtin_nontemporal_load(mrow + cI);   // single-use: NT load
            t = (m == 0) ? -1e20f : t;
            t *= SCALE;                       // reference: (mask ? att : -1e20)/sqrt(128)
            srow[cI] = t;
            mx = fmaxf(mx, t);
        }
#pragma unroll
        for (int off = 16; off >= 1; off >>= 1)
            mx = fmaxf(mx, __shfl_xor(mx, off, 32));

        float sum = 0.f;
        for (int cI = lane; cI < SEQ; cI += 32) {
            float e = __expf(srow[cI] - mx);
            srow[cI] = e;
            sum += e;
        }
#pragma unroll
        for (int off = 16; off >= 1; off >>= 1)
            sum += __shfl_xor(sum, off, 32);
        const float inv = 1.f / sum;

        float* arow = attw + (((size_t)(n * HEADS + h)) * SEQ + grow) * SEQ;
        for (int cI = lane; cI < SEQ; cI += 32) {
            const float w = srow[cI] * inv;
            srow[cI] = w;                                 // kept for phase 3
            __builtin_nontemporal_store(w, arow + cI);    // streaming output: NT store
        }
    }
    __syncthreads();

    // ---- phase 3: head_out(16x32) = attw @ V; each wave owns a 128-wide K slice ----
    const int kbase = wv * 128;
#pragma unroll
    for (int t = 0; t < 2; ++t) {                 // two 16-wide D column tiles
        v8f c = {};
#pragma unroll 4
        for (int kc = 0; kc < 32; ++kc) {
            const int kk = kbase + kc * 4 + khalf;
            v2f a2, b2;
            a2.x = scores[lm * SEQ + kk];
            a2.y = scores[lm * SEQ + kk + 1];
            b2.x = vb[(size_t)kk * HD + t * 16 + lm];
            b2.y = vb[(size_t)(kk + 1) * HD + t * 16 + lm];
            c = __builtin_amdgcn_wmma_f32_16x16x4_f32(false, a2, false, b2,
                                                      (short)0, c, false, false);
        }
#pragma unroll
        for (int r = 0; r < 8; ++r)               // merge wave partials: ds_add_f32
            atomicAdd(&accum[(r + (khalf << 2)) * HD + t * 16 + lm], c[r]);
    }
    __syncthreads();

    // ---- phase 4: write head output in [N,S,E] layout for the final GEMM ----
    for (int e = tid; e < 16 * HD; e += 256) {
        const int m = e >> 5, d = e & 31;
        attout[((size_t)n * SEQ + row0 + m) * EMBED + h * HD + d] = accum[e];
    }
}

// ---------------------------------------------------------------------------
// Kernel C: out = attout @ Wo^T + bo   (small FLOP share; Wo stays L2-resident)
// ---------------------------------------------------------------------------
__global__ void out_proj_kernel(const float* __restrict__ attout,
                                const float* __restrict__ Wo,
                                const float* __restrict__ bo,
                                float* __restrict__ out)
{
    __shared__ float sx[16 * EMBED];   // 16 tokens, 8 KB
    const int tid    = threadIdx.x;    // 256
    const int token0 = blockIdx.x * 16;
    for (int i = tid; i < 16 * EMBED; i += 256)
        sx[i] = attout[(size_t)token0 * EMBED + i];
    __syncthreads();

#pragma unroll
    for (int it = 0; it < 8; ++it) {
        const int idx = it * 256 + tid;
        const int tok = idx >> 7;
        const int e   = idx & 127;
        float acc = bo[e];
        const float* wrow = Wo + (size_t)e * EMBED;
        const float* xrow = sx + tok * EMBED;
#pragma unroll 8
        for (int j = 0; j < EMBED; ++j)
            acc += xrow[j] * wrow[j];
        out[(size_t)(token0 + tok) * EMBED + e] = acc;
    }
}

// ---------------------------------------------------------------------------
extern "C" void kernel_launch(void* const* d_in, const int* in_sizes, int n_in,
                              void* d_out, int out_size, void* d_ws, size_t ws_size,
                              hipStream_t stream)
{
    const float* values = (const float*)d_in[0];
    const float* keys_  = (const float*)d_in[1];
    const float* query  = (const float*)d_in[2];
    const int*   mask   = (const int*)d_in[3];
    const float* Wv     = (const float*)d_in[4];
    const float* Wk     = (const float*)d_in[5];
    const float* Wq     = (const float*)d_in[6];
    const float* Wo     = (const float*)d_in[7];
    const float* bo     = (const float*)d_in[8];

    float* ws = (float*)d_ws;
    const size_t perT = (size_t)N_BATCH * HEADS * SEQ * HD;  // 4,194,304 floats
    float* qw     = ws;
    float* kw     = ws +     perT;
    float* vw     = ws + 2 * perT;
    float* attout = ws + 3 * perT;                            // 64 MB total ws use

    float* out  = (float*)d_out;
    float* attw = out + (size_t)N_BATCH * SEQ * EMBED;        // 2nd output (tuple)

    qkv_proj_kernel<<<N_BATCH * SEQ, 128, 0, stream>>>(
        values, keys_, query, Wv, Wk, Wq, qw, kw, vw);

    const size_t smemB = (size_t)(16 * SEQ + 16 * HD) * sizeof(float);  // 67,584 B (<320 KB WGP LDS)
    attention_kernel<<<dim3(SEQ / 16, HEADS, N_BATCH), 256, smemB, stream>>>(
        qw, kw, vw, mask, attw, attout);

    out_proj_kernel<<<(N_BATCH * SEQ) / 16, 256, 0, stream>>>(attout, Wo, bo, out);
}